// BitFlipLinear_89404039233680
// MI455X (gfx1250) — compile-verified
//
#include <hip/hip_runtime.h>

// Y[M,N] = X[M,K] * W[N,K]^T + bias[N]
// M = 32768, N = 4096, K = 4096, fp32 end-to-end.
// Pipeline: double-buffered LDS tiles staged with GLOBAL_LOAD_ASYNC_TO_LDS_B128
// (ASYNCcnt), compute with V_WMMA_F32_16X16X4_F32.

#define M_TOTAL 32768
#define D_N 4096
#define D_K 4096

#define BM 128
#define BN 128
#define BK 32
#define NT (D_K / BK)     // 128 K-tiles
#define LDKF 36           // padded f32 LDS row stride: 144 B (16B aligned)

typedef __attribute__((ext_vector_type(2))) float v2f;
typedef __attribute__((ext_vector_type(4))) float v4f;
typedef __attribute__((ext_vector_type(8))) float v8f;
typedef __attribute__((ext_vector_type(4))) int   v4i;

#if __has_builtin(__builtin_amdgcn_global_load_async_to_lds_b128)
#define HAVE_ASYNC 1
#else
#define HAVE_ASYNC 0
#endif

__device__ __forceinline__ void stage_b128(const float* g, float* l) {
#if HAVE_ASYNC
    __builtin_amdgcn_global_load_async_to_lds_b128(
        (__attribute__((address_space(1))) v4i*)(v4i*)(void*)g,
        (__attribute__((address_space(3))) v4i*)(v4i*)(void*)l,
        /*offset=*/0, /*cpol=*/0);
#else
    *(v4f*)l = *(const v4f*)g;   // synchronous fallback, same pipeline point
#endif
}

template <int N>
__device__ __forceinline__ void wait_async() {
#if HAVE_ASYNC
#if __has_builtin(__builtin_amdgcn_s_wait_asynccnt)
    __builtin_amdgcn_s_wait_asynccnt(N);
#else
    asm volatile("s_wait_asynccnt %0" ::"i"(N) : "memory");
#endif
#endif
}

__global__ __launch_bounds__(256) void ternary_linear_wmma_f32(
    const float* __restrict__ X, const float* __restrict__ W,
    const float* __restrict__ bias, float* __restrict__ Y)
{
    __shared__ float As[2][BM][LDKF];   // X tiles  [buf][m][k]
    __shared__ float Bs[2][BN][LDKF];   // W tiles  [buf][n][k]

    const int tid  = threadIdx.x;
    const int lane = tid & 31;
    const int wave = tid >> 5;
    const int wm   = wave >> 2;         // 0..1 -> 64-row slab
    const int wn   = wave & 3;          // 0..3 -> 32-col slab

    const int m0 = blockIdx.y * BM;
    const int n0 = blockIdx.x * BN;

    // ---- per-thread staging slots: 4 x 16B chunks per matrix ----
    int srow[4], skq[4];
#pragma unroll
    for (int i = 0; i < 4; ++i) {
        const int f = tid + i * 256;    // float4 slot 0..1023
        srow[i] = f >> 3;               // 0..127
        skq[i]  = (f & 7) * 4;          // 0,4,...,28
    }

    auto issue_tile = [&](int t, int buf) {
#pragma unroll
        for (int i = 0; i < 4; ++i) {
            const float* gx = X + (size_t)(m0 + srow[i]) * D_K + t * BK + skq[i];
            const float* gw = W + (size_t)(n0 + srow[i]) * D_K + t * BK + skq[i];
            stage_b128(gx, &As[buf][srow[i]][skq[i]]);
            stage_b128(gw, &Bs[buf][srow[i]][skq[i]]);
        }
    };

    v8f acc[4][2];
#pragma unroll
    for (int mi = 0; mi < 4; ++mi)
#pragma unroll
        for (int ni = 0; ni < 2; ++ni)
#pragma unroll
            for (int e = 0; e < 8; ++e)
                acc[mi][ni][e] = 0.0f;

    const int lrow = lane & 15;         // A: M row / B: N col within 16
    const int kk2  = (lane >> 4) * 2;   // K pair base within a K=4 step

    issue_tile(0, 0);                   // prologue: tile 0 -> buf 0

    for (int t = 0; t < NT; ++t) {
        const int p = t & 1;

        if (t + 1 < NT) {
            issue_tile(t + 1, 1 - p);   // buf[1-p] reads finished last iter
            wait_async<8>();            // 8 newer ops may stay outstanding;
                                        // in-order retire => tile t complete
        } else {
            wait_async<0>();
        }
        __syncthreads();                // tile t visible to all waves

        const float (*Ab)[LDKF] = As[p];
        const float (*Bb)[LDKF] = Bs[p];

#pragma unroll
        for (int ks = 0; ks < BK / 4; ++ks) {
            v2f a[4], b[2];
#pragma unroll
            for (int mi = 0; mi < 4; ++mi)
                a[mi] = *(const v2f*)(&Ab[wm * 64 + mi * 16 + lrow][ks * 4 + kk2]);
#pragma unroll
            for (int ni = 0; ni < 2; ++ni)
                b[ni] = *(const v2f*)(&Bb[wn * 32 + ni * 16 + lrow][ks * 4 + kk2]);

#pragma unroll
            for (int mi = 0; mi < 4; ++mi)
#pragma unroll
                for (int ni = 0; ni < 2; ++ni)
                    acc[mi][ni] = __builtin_amdgcn_wmma_f32_16x16x4_f32(
                        /*neg_a=*/false, a[mi],
                        /*neg_b=*/false, b[ni],
                        /*c_mod=*/(short)0, acc[mi][ni],
                        /*reuse_a=*/false, /*reuse_b=*/false);
        }

        __syncthreads();                // all waves done reading buf[p]
    }

    // ---- epilogue: add bias, store fp32 ----
#pragma unroll
    for (int ni = 0; ni < 2; ++ni) {
        const int gn = n0 + wn * 32 + ni * 16 + lrow;   // D: lane&15 = N col
        const float bv = bias[gn];
#pragma unroll
        for (int mi = 0; mi < 4; ++mi) {
            const int mbase = m0 + wm * 64 + mi * 16 + (lane >> 4) * 8;
#pragma unroll
            for (int r = 0; r < 8; ++r) {
                Y[(size_t)(mbase + r) * D_N + gn] = acc[mi][ni][r] + bv;
            }
        }
    }
}

extern "C" void kernel_launch(void* const* d_in, const int* in_sizes, int n_in,
                              void* d_out, int out_size, void* d_ws, size_t ws_size,
                              hipStream_t stream) {
    const float* x    = (const float*)d_in[0];   // [B,S,K] = [M,K]
    const float* w    = (const float*)d_in[1];   // [N,K]
    const float* bias = (const float*)d_in[2];   // [N]
    float* y          = (float*)d_out;           // [M,N]

    dim3 grid(D_N / BN, M_TOTAL / BM);           // N blocks fastest -> L2 reuse of X strip
    ternary_linear_wmma_f32<<<grid, dim3(256), 0, stream>>>(x, w, bias, y);
}